// GraphConv_57363583205766
// MI455X (gfx1250) — compile-verified
//
#include <hip/hip_runtime.h>

// GraphConv scatter: out[t] += inputs[s] * (esgn*enorm), per edge.
// N=100000, E=1600000, D=64 (f32). Inputs (25.6MB), output (25.6MB) and edge
// data (25.6MB) are all L2-resident on MI455X (192MB L2) -> the kernel is
// L2-gather + f32-atomic bound, not HBM bound. One wave32 per edge:
// 32 lanes x float2 = one 256B feature row per VMEM request.

#define TILE 256   // edges staged per 256-thread block (8 waves x 32 edges)

typedef __attribute__((address_space(1))) int  glb_i32;   // global (AS1)
typedef __attribute__((address_space(3))) int  lds_i32;   // LDS    (AS3)

struct alignas(16) EdgeRec {   // one 16B record -> single ds_load_b128 broadcast
    int   s;
    int   t;
    float n;
    float g;
};

__device__ __forceinline__ void wait_asynccnt0() {
#if __has_builtin(__builtin_amdgcn_s_wait_asynccnt)
    __builtin_amdgcn_s_wait_asynccnt(0);
#else
    asm volatile("s_wait_asynccnt 0" ::: "memory");
#endif
}

__global__ __launch_bounds__(256) void graphconv_scatter(
    const float* __restrict__ inputs,   // [N, 64]
    const int*   __restrict__ sidx,     // [E]
    const int*   __restrict__ tidx,     // [E]
    const float* __restrict__ enorm,    // [E]
    const float* __restrict__ esgn,     // [E]
    float*       __restrict__ out,      // [N, 64]
    int n_edges)
{
    __shared__ EdgeRec s_rec[TILE];     // 4KB of 320KB LDS -> high occupancy

    const int tid  = threadIdx.x;
    const int base = blockIdx.x * TILE;
    const int e    = base + tid;

    // ---- Stage this block's 256 edge records into LDS (coalesced) ----
#if defined(__gfx1250__) && __has_builtin(__builtin_amdgcn_global_load_async_to_lds_b32)
    if (e < n_edges) {
        __builtin_amdgcn_global_load_async_to_lds_b32(
            (glb_i32*)(sidx + e),  (lds_i32*)&s_rec[tid].s, 0, 0);
        __builtin_amdgcn_global_load_async_to_lds_b32(
            (glb_i32*)(tidx + e),  (lds_i32*)&s_rec[tid].t, 0, 0);
        __builtin_amdgcn_global_load_async_to_lds_b32(
            (glb_i32*)(enorm + e), (lds_i32*)&s_rec[tid].n, 0, 0);
        __builtin_amdgcn_global_load_async_to_lds_b32(
            (glb_i32*)(esgn + e),  (lds_i32*)&s_rec[tid].g, 0, 0);
    }
    wait_asynccnt0();          // each wave drains its own ASYNCcnt
    __syncthreads();           // then all waves' LDS writes are visible
#else
    if (e < n_edges) {
        EdgeRec r;
        r.s = sidx[e];
        r.t = tidx[e];
        r.n = enorm[e];
        r.g = esgn[e];
        s_rec[tid] = r;
    }
    __syncthreads();
#endif

    const int lane  = tid & 31;
    const int wave  = tid >> 5;
    const int count = (n_edges - base < TILE) ? (n_edges - base) : TILE;

    // ---- Each wave processes 32 staged edges: gather row, scale, atomic-scatter ----
    int lo = wave * 32;
    int hi = lo + 32;
    if (hi > count) hi = count;

#pragma unroll 8
    for (int i = lo; i < hi; ++i) {
        const EdgeRec r = s_rec[i];          // one ds_load_b128, broadcast to all lanes
        const float   w = r.n * r.g;

        const float2* src = (const float2*)(inputs + (size_t)r.s * 64);
        float2 x = src[lane];                // global_load_b64: 32 lanes = full 256B row

        float* dst = out + (size_t)r.t * 64 + lane * 2;
        // Hardware f32 atomics, no return value -> non-returning form (STOREcnt only)
        unsafeAtomicAdd(dst,     x.x * w);
        unsafeAtomicAdd(dst + 1, x.y * w);
    }
}

__global__ __launch_bounds__(256) void graphconv_zero(float4* __restrict__ out, int n4) {
    int i = blockIdx.x * blockDim.x + threadIdx.x;
    if (i < n4) out[i] = make_float4(0.f, 0.f, 0.f, 0.f);
}

extern "C" void kernel_launch(void* const* d_in, const int* in_sizes, int n_in,
                              void* d_out, int out_size, void* d_ws, size_t ws_size,
                              hipStream_t stream) {
    const float* inputs = (const float*)d_in[0];   // [N, 64]
    const int*   eidx   = (const int*)d_in[1];     // [2, E]
    const float* enorm  = (const float*)d_in[2];   // [E]
    const float* esgn   = (const float*)d_in[3];   // [E]
    float*       out    = (float*)d_out;           // [N, 64]

    const int E  = in_sizes[1] / 2;
    const int n4 = out_size / 4;                   // out_size = N*64, divisible by 4

    // Output is accumulated into -> must be zeroed every call (harness poisons it).
    graphconv_zero<<<(n4 + 255) / 256, 256, 0, stream>>>((float4*)out, n4);

    const int blocks = (E + TILE - 1) / TILE;
    graphconv_scatter<<<blocks, 256, 0, stream>>>(inputs, eidx, eidx + E,
                                                  enorm, esgn, out, E);
}